// SparseAutoencoder_51711406244195
// MI455X (gfx1250) — compile-verified
//
#include <hip/hip_runtime.h>
#include <math.h>

#define D_IN   1024
#define LAT    8192
#define NROWS  16384
#define TOPK_K 64

typedef __attribute__((ext_vector_type(2))) float        v2f;
typedef __attribute__((ext_vector_type(8))) float        v8f;
typedef __attribute__((ext_vector_type(4))) unsigned int u32x4;
typedef __attribute__((ext_vector_type(4))) int          i32x4;
typedef __attribute__((ext_vector_type(8))) int          i32x8;

#ifndef __has_builtin
#define __has_builtin(x) 0
#endif
#if defined(__gfx1250__) && __has_builtin(__builtin_amdgcn_tensor_load_to_lds) && \
    __has_builtin(__builtin_amdgcn_s_wait_tensorcnt)
#define SAE_USE_TDM 1
#else
#define SAE_USE_TDM 0
#endif

// ---------------------------------------------------------------------------
// Kernel 1: W = prev + alpha*(dec - prev); also write transposed Wt to ws.
// alpha = (task_id==0) ? 1 : sigmoid(s)/(task_id+1)   (computed on device)
// ---------------------------------------------------------------------------
__global__ __launch_bounds__(256)
void sae_make_w_kernel(const float* __restrict__ dec_w,
                       const float* __restrict__ prev_w,
                       const float* __restrict__ s_ptr,
                       const int*   __restrict__ task_ptr,
                       float* __restrict__ W_out,
                       float* __restrict__ Wt,
                       int writeWt)
{
    __shared__ float tile[32][33];
    const int jt = blockIdx.x * 32;   // latent (col) tile
    const int dt = blockIdx.y * 32;   // input-dim (row) tile
    const float s = s_ptr[0];
    const int task = task_ptr[0];
    const float sig = 1.0f / (1.0f + __expf(-s));
    const float alpha = (task == 0) ? 1.0f : sig / (float)(task + 1);

    const int lx = threadIdx.x;       // 0..31 -> j within tile
    const int ly = threadIdx.y;       // 0..7

    #pragma unroll
    for (int r = 0; r < 4; ++r) {
        const int dd = dt + ly + r * 8;
        const int jj = jt + lx;
        const size_t off = (size_t)dd * LAT + jj;
        const float pw = prev_w[off];
        const float w  = pw + alpha * (dec_w[off] - pw);
        W_out[off] = w;                       // coalesced along j
        tile[ly + r * 8][lx] = w;             // tile[d_local][j_local]
    }
    __syncthreads();
    if (writeWt) {
        #pragma unroll
        for (int r = 0; r < 4; ++r) {
            const int jj = jt + ly + r * 8;
            const int dd = dt + lx;
            Wt[(size_t)jj * D_IN + dd] = tile[lx][ly + r * 8];  // coalesced along d
        }
    }
}

// ---------------------------------------------------------------------------
// TDM helper: DMA a 2-D tile of 128 rows x 32 f32 (row stride D_IN) from
// global into LDS, inserting 4 DWORDs of pad after every 32 DWORDs so the
// LDS layout is [row][36] floats (bank-conflict-free fragment reads).
// Descriptor per CDNA5 ISA (D# groups 0/1; 2D -> g2=g3=0).
// This toolchain's builtin is the 6-arg form:
//   (u32x4 g0, i32x8 g1, i32x4 g2, i32x4 g3, i32x8 extra, i32 cpol)
// ---------------------------------------------------------------------------
#if SAE_USE_TDM
__device__ __forceinline__ unsigned sae_lds_off(const void* p)
{
    return (unsigned)(unsigned long long)(const __attribute__((address_space(3))) char*)p;
}

__device__ __forceinline__ void sae_tdm_load_tile(const float* gsrc,
                                                  unsigned lds_off,
                                                  unsigned rows_avail)
{
    const unsigned long long ga = (unsigned long long)gsrc;
    u32x4 g0;
    g0.x = 1u;                                           // count=1, user mode
    g0.y = lds_off;                                      // lds_addr (bytes)
    g0.z = (unsigned)(ga & 0xFFFFFFFFu);                 // global_addr lo
    g0.w = (unsigned)((ga >> 32) & 0x01FFFFFFu)          // global_addr hi (57b)
         | (2u << 30);                                   // type = 2 ("image")

    i32x8 g1;
    g1[0] = (int)((2u << 16)      // data_size = 2 -> 4 bytes
                | (1u << 20)      // pad_enable
                | (4u << 22)      // pad_interval = 4 -> every 32 DWORDs
                | (3u << 25));    // pad_amount   = 3 -> 4 DWORDs
    g1[1] = (int)(((unsigned)D_IN & 0xFFFFu) << 16);     // tensor_dim0[15:0]
    g1[2] = (int)((((unsigned)D_IN >> 16) & 0xFFFFu)     // tensor_dim0[31:16]
                | ((rows_avail & 0xFFFFu) << 16));       // tensor_dim1[15:0]
    g1[3] = (int)(((rows_avail >> 16) & 0xFFFFu)         // tensor_dim1[31:16]
                | (32u << 16));                          // tile_dim0 = 32
    g1[4] = (int)128;                                    // tile_dim1 = 128
    g1[5] = (int)D_IN;                                   // tensor_dim0_stride lo
    g1[6] = 0;
    g1[7] = 0;

    i32x4 z4 = {0, 0, 0, 0};
    i32x8 z8 = {0, 0, 0, 0, 0, 0, 0, 0};
    __builtin_amdgcn_tensor_load_to_lds(g0, g1, z4, z4, z8, 0);
}
#endif

// ---------------------------------------------------------------------------
// Kernel 2: latent = x @ enc_w^T + enc_b, fp32 WMMA 16x16x4.
// Block tile 128(M) x 128(N) x 32(K); 8 waves, wave tile 64x32 (4x2 frags).
// Double-buffered LDS tiles [row][36] staged by TDM (fallback: float4 copy).
// A frag (16x4 f32): lanes 0-15 hold {K,K+1}, lanes 16-31 hold {K+2,K+3}.
// ---------------------------------------------------------------------------
__global__ __launch_bounds__(256)
void sae_encoder_wmma_kernel(const float* __restrict__ x,
                             const float* __restrict__ enc_w,
                             const float* __restrict__ enc_b,
                             float* __restrict__ latent)
{
    constexpr int BM = 128, BN = 128, BK = 32;
    constexpr int LDR = BK + 4;                         // padded row stride
    __shared__ __align__(16) float lds[2][2][BM * LDR]; // [buf][A=0/B=1]

    const int tid  = threadIdx.x;
    const int lane = tid & 31;
    const int wave = tid >> 5;
    const int wm   = wave >> 2;                         // 0..1 (M)
    const int wn   = wave & 3;                          // 0..3 (N)
    const int lh   = lane >> 4;
    const int ll   = lane & 15;

    const int rowB = blockIdx.y * BM;
    const int colB = blockIdx.x * BN;

    const float* gA = x     + (size_t)rowB * D_IN;
    const float* gB = enc_w + (size_t)colB * D_IN;

    v8f acc[4][2] = {};

#if SAE_USE_TDM
    const unsigned offA[2] = { sae_lds_off(&lds[0][0][0]), sae_lds_off(&lds[1][0][0]) };
    const unsigned offB[2] = { sae_lds_off(&lds[0][1][0]), sae_lds_off(&lds[1][1][0]) };
    if (wave == 0) {
        sae_tdm_load_tile(gA, offA[0], (unsigned)(NROWS - rowB));
        sae_tdm_load_tile(gB, offB[0], (unsigned)(LAT - colB));
    }
#else
    const int ldm = tid >> 3;                           // 0..31 (row)
    const int ldk = (tid & 7) * 4;                      // 0..28 (k)
    #pragma unroll
    for (int r = 0; r < 4; ++r) {
        const int m = ldm + r * 32;
        *(float4*)(&lds[0][0][m * LDR + ldk]) = *(const float4*)(gA + (size_t)m * D_IN + ldk);
        *(float4*)(&lds[0][1][m * LDR + ldk]) = *(const float4*)(gB + (size_t)m * D_IN + ldk);
    }
#endif

    constexpr int NIT = D_IN / BK;                      // 32 K-steps
    for (int it = 0; it < NIT; ++it) {
        const int buf = it & 1;
#if SAE_USE_TDM
        if (wave == 0) __builtin_amdgcn_s_wait_tensorcnt(0);
#endif
        __syncthreads();    // tile[buf] ready; prev compute on buf^1 done

        // prefetch next K-step into the other buffer while computing
#if SAE_USE_TDM
        if (it + 1 < NIT && wave == 0) {
            const int kb = (it + 1) * BK;
            sae_tdm_load_tile(gA + kb, offA[buf ^ 1], (unsigned)(NROWS - rowB));
            sae_tdm_load_tile(gB + kb, offB[buf ^ 1], (unsigned)(LAT - colB));
        }
#else
        if (it + 1 < NIT) {
            const int kb = (it + 1) * BK;
            #pragma unroll
            for (int r = 0; r < 4; ++r) {
                const int m = ldm + r * 32;
                *(float4*)(&lds[buf ^ 1][0][m * LDR + ldk]) =
                    *(const float4*)(gA + (size_t)m * D_IN + kb + ldk);
                *(float4*)(&lds[buf ^ 1][1][m * LDR + ldk]) =
                    *(const float4*)(gB + (size_t)m * D_IN + kb + ldk);
            }
        }
#endif
        const float* As = &lds[buf][0][0];
        const float* Bs = &lds[buf][1][0];
        #pragma unroll
        for (int k4 = 0; k4 < BK; k4 += 4) {
            v2f a[4], b[2];
            #pragma unroll
            for (int mf = 0; mf < 4; ++mf) {
                const int m = wm * 64 + mf * 16 + ll;
                a[mf] = *(const v2f*)(As + m * LDR + k4 + 2 * lh);   // ds_load_b64
            }
            #pragma unroll
            for (int nf = 0; nf < 2; ++nf) {
                const int n = wn * 32 + nf * 16 + ll;
                b[nf] = *(const v2f*)(Bs + n * LDR + k4 + 2 * lh);   // ds_load_b64
            }
            #pragma unroll
            for (int mf = 0; mf < 4; ++mf)
                #pragma unroll
                for (int nf = 0; nf < 2; ++nf)
                    acc[mf][nf] = __builtin_amdgcn_wmma_f32_16x16x4_f32(
                        false, a[mf], false, b[nf], (short)0, acc[mf][nf],
                        false, false);
        }
    }

    // ---- epilogue: bias + store (C/D layout: vgpr r -> M=r, +8 upper half) ----
    #pragma unroll
    for (int nf = 0; nf < 2; ++nf) {
        const int n = colB + wn * 32 + nf * 16 + ll;
        const float bias = enc_b[n];
        #pragma unroll
        for (int mf = 0; mf < 4; ++mf) {
            const int mBase = rowB + wm * 64 + mf * 16 + lh * 8;
            #pragma unroll
            for (int r = 0; r < 8; ++r) {
                latent[(size_t)(mBase + r) * LAT + n] = acc[mf][nf][r] + bias;
            }
        }
    }
}

// ---------------------------------------------------------------------------
// Kernel 3: per-row fused top-64 (exact radix select on |v| bits) + sparse
// decoder mod[row] = sum_k val_k * W[:, idx_k].  One 256-thread WG per row.
// USE_WT=1: Wsrc is Wt[LAT][D_IN] (coalesced);  USE_WT=0: Wsrc is W[D_IN][LAT].
// ---------------------------------------------------------------------------
template <int USE_WT>
__global__ __launch_bounds__(256)
void sae_topk_decode_kernel(const float* __restrict__ latent,
                            const float* __restrict__ Wsrc,
                            float* __restrict__ mod)
{
    __shared__ __align__(16) float s_row[LAT];     // 32 KB row in LDS
    __shared__ unsigned s_scan[257];               // hist / suffix-scan
    __shared__ unsigned s_meta[4];                 // prefix, remk, cnt_gt, cnt_eq
    __shared__ int      s_idx[TOPK_K];
    __shared__ float    s_val[TOPK_K];

    const int tid = threadIdx.x;
    const size_t row = blockIdx.x;
    const float* lrow = latent + row * LAT;

    // load row (fully coalesced float4)
    {
        const float4* g4 = (const float4*)lrow;
        float4* l4 = (float4*)s_row;
        #pragma unroll
        for (int t = 0; t < 8; ++t) l4[t * 256 + tid] = g4[t * 256 + tid];
    }
    if (tid == 0) { s_meta[0] = 0u; s_meta[1] = (unsigned)TOPK_K; }
    __syncthreads();

    // ---- 4-pass radix select (MSB-first) on abs-value bit pattern ----
    for (int pass = 3; pass >= 0; --pass) {
        const int shift = pass * 8;
        const unsigned prefix = s_meta[0];
        const unsigned remk   = s_meta[1];
        const unsigned maskDone = (pass == 3) ? 0u : (0xFFFFFFFFu << (shift + 8));

        s_scan[tid] = 0u;
        if (tid == 0) s_scan[256] = 0u;
        __syncthreads();

        #pragma unroll 4
        for (int t = 0; t < 32; ++t) {
            const unsigned k = __float_as_uint(s_row[t * 256 + tid]) & 0x7FFFFFFFu;
            if ((k & maskDone) == prefix)
                atomicAdd(&s_scan[(k >> shift) & 0xFFu], 1u);
        }
        __syncthreads();

        // suffix sum: s_scan[b] = #candidates with digit >= b
        for (int off = 1; off < 256; off <<= 1) {
            const unsigned v = s_scan[tid];
            const unsigned w = (tid + off < 256) ? s_scan[tid + off] : 0u;
            __syncthreads();
            s_scan[tid] = v + w;
            __syncthreads();
        }

        const unsigned ge = s_scan[tid];
        const unsigned gt = s_scan[tid + 1];   // tid==255 -> s_scan[256]==0
        if (ge >= remk && gt < remk) {         // unique digit bin
            s_meta[0] = prefix | ((unsigned)tid << shift);
            s_meta[1] = remk - gt;
        }
        __syncthreads();
    }

    const unsigned T    = s_meta[0];   // exact key of 64th-largest |v|
    const unsigned remk = s_meta[1];   // # of ==T elements to take
    if (tid == 0) { s_meta[2] = 0u; s_meta[3] = 0u; }
    __syncthreads();

    // gather strictly-greater elements
    #pragma unroll 4
    for (int t = 0; t < 32; ++t) {
        const int j = t * 256 + tid;
        const float v = s_row[j];
        const unsigned k = __float_as_uint(v) & 0x7FFFFFFFu;
        if (k > T) {
            const unsigned p = atomicAdd(&s_meta[2], 1u);
            if (p < TOPK_K) { s_idx[p] = j; s_val[p] = v; }
        }
    }
    __syncthreads();
    const unsigned cgt = s_meta[2];
    // fill remaining slots with ==T elements
    #pragma unroll 4
    for (int t = 0; t < 32; ++t) {
        const int j = t * 256 + tid;
        const float v = s_row[j];
        const unsigned k = __float_as_uint(v) & 0x7FFFFFFFu;
        if (k == T) {
            const unsigned p = atomicAdd(&s_meta[3], 1u);
            if (p < remk && cgt + p < TOPK_K) { s_idx[cgt + p] = j; s_val[cgt + p] = v; }
        }
    }
    __syncthreads();

    // ---- sparse decode: 64-column rank update, 4 output dims per thread ----
    float acc0 = 0.f, acc1 = 0.f, acc2 = 0.f, acc3 = 0.f;
    #pragma unroll 8
    for (int k = 0; k < TOPK_K; ++k) {
        const int j = s_idx[k];       // LDS broadcast
        const float v = s_val[k];
        if (USE_WT) {
            const float* wr = Wsrc + (size_t)j * D_IN;   // coalesced L2-resident row
            acc0 += v * wr[tid];
            acc1 += v * wr[tid + 256];
            acc2 += v * wr[tid + 512];
            acc3 += v * wr[tid + 768];
        } else {
            acc0 += v * Wsrc[(size_t)(tid)       * LAT + j];
            acc1 += v * Wsrc[(size_t)(tid + 256) * LAT + j];
            acc2 += v * Wsrc[(size_t)(tid + 512) * LAT + j];
            acc3 += v * Wsrc[(size_t)(tid + 768) * LAT + j];
        }
    }
    float* out = mod + row * D_IN;
    out[tid]       = acc0;
    out[tid + 256] = acc1;
    out[tid + 512] = acc2;
    out[tid + 768] = acc3;
}

// ---------------------------------------------------------------------------
extern "C" void kernel_launch(void* const* d_in, const int* in_sizes, int n_in,
                              void* d_out, int out_size, void* d_ws, size_t ws_size,
                              hipStream_t stream)
{
    const float* x       = (const float*)d_in[0];  // [8,2048,1024]
    const float* enc_w   = (const float*)d_in[1];  // [8192,1024]
    const float* enc_b   = (const float*)d_in[2];  // [8192]
    const float* dec_w   = (const float*)d_in[3];  // [1024,8192]
    const float* s       = (const float*)d_in[4];  // scalar
    const float* prev_w  = (const float*)d_in[5];  // [1024,8192]
    const int*   task_id = (const int*)d_in[6];    // scalar

    float* mod    = (float*)d_out;                       // 16384*1024
    float* latent = mod + (size_t)NROWS * D_IN;          // 16384*8192
    float* W      = latent + (size_t)NROWS * LAT;        // 1024*8192

    float* Wt = (float*)d_ws;
    const bool useWt = ws_size >= (size_t)D_IN * LAT * sizeof(float);

    // 1) interpolated decoder weight (+ transpose into workspace)
    {
        dim3 blk(32, 8);
        dim3 grd(LAT / 32, D_IN / 32);
        sae_make_w_kernel<<<grd, blk, 0, stream>>>(dec_w, prev_w, s, task_id,
                                                   W, Wt, useWt ? 1 : 0);
    }
    // 2) encoder GEMM (fp32 WMMA, TDM-staged double-buffered tiles)
    {
        dim3 grd(LAT / 128, NROWS / 128);   // (64, 128)
        sae_encoder_wmma_kernel<<<grd, 256, 0, stream>>>(x, enc_w, enc_b, latent);
    }
    // 3) fused exact top-64 + sparse decoder
    if (useWt)
        sae_topk_decode_kernel<1><<<NROWS, 256, 0, stream>>>(latent, Wt, mod);
    else
        sae_topk_decode_kernel<0><<<NROWS, 256, 0, stream>>>(latent, W, mod);
}